// StandardRasterizer_33174327394987
// MI455X (gfx1250) — compile-verified
//
#include <hip/hip_runtime.h>
#include <stdint.h>

// Geometry from the reference: B=4, F=10000, D=16, H=W=512.
constexpr int TPB    = 256;
constexpr int ITERS  = 4;
constexpr int BLOCKS = 1024;              // TPB*ITERS*BLOCKS == 2^20 == B*H*W
constexpr int HWPIX  = 512 * 512;         // 262144 == 2^18
constexpr int DCH    = 16;

constexpr int BARY_BYTES = TPB * 12;      // 3072 B of bary per tile
constexpr int TILE_BYTES = TPB * 16;      // 4096 B per tile: [0,3072) bary, [3072,4096) tri
constexpr int BARY_CHUNKS = BARY_BYTES / 16;  // 192 == 6 waves worth -> no wave diverges

// Low 32 bits of a flat pointer to a __shared__ object == LDS byte offset
// (flat shared addr = {SHARED_BASE[15:0],16'b0} << 32 | offset, ISA 10.2).
__device__ __forceinline__ uint32_t lds_off_u32(const void* p) {
    return (uint32_t)(uintptr_t)p;
}

// CDNA5 async DMA: global -> LDS (per-lane addressed, tracked with ASYNCcnt).
__device__ __forceinline__ void async_load_b128(uint32_t lds, uint64_t gaddr) {
    asm volatile("global_load_async_to_lds_b128 %0, %1, off"
                 :: "v"(lds), "v"(gaddr) : "memory");
}

__global__ __launch_bounds__(TPB)
void raster_interp_kernel(const float* __restrict__ attributes,   // [B*F, 3, 16]
                          const int*   __restrict__ tribuf,       // [B, H, W]
                          const float* __restrict__ barybuf,      // [B, H, W, 3]
                          float*       __restrict__ out)          // [B, 17, H, W]
{
    // Double-buffered staging of the streaming inputs: one async b128 per
    // wave per tile stages the whole 4 KB block (bary + tri) cooperatively.
    __shared__ uint8_t stage[2][TILE_BYTES];

    const int tid   = threadIdx.x;
    const int span  = BLOCKS * TPB;                   // pixels per sweep
    const int pbase = blockIdx.x * TPB;               // this block's tile base

    // Thread t stages 16-byte chunk t: threads [0,192) -> bary, [192,256) -> tri.
    // Both tile bases are computed uniformly; the tri base is pre-rebased by
    // -192*16 so the per-lane offset is tid*16 on BOTH sides. The select is
    // then a plain 64-bit v_cndmask (no saveexec/branch divergence).
    auto issue_tile = [&](int t, int buf) {
        const uint64_t tpix  = (uint64_t)t * span + pbase;   // tile's first pixel
        const uint64_t baryT = (uint64_t)(uintptr_t)barybuf + tpix * 12;
        const uint64_t triT  = (uint64_t)(uintptr_t)tribuf  + tpix * 4
                               - (uint64_t)BARY_CHUNKS * 16;
        const uint64_t base  = (tid < BARY_CHUNKS) ? baryT : triT;
        async_load_b128(lds_off_u32(&stage[buf][0]) + tid * 16,
                        base + (uint64_t)tid * 16);
    };

    issue_tile(0, 0);                                 // prime (ASYNCcnt = 1 per wave)

    for (int t = 0; t < ITERS; ++t) {
        const int cur = t & 1;

        // My wave's tile-t chunk is in LDS...
        asm volatile("s_wait_asynccnt 0" ::: "memory");
        // ...and after the barrier, every wave's chunk is. The barrier also
        // proves all waves finished reading buf cur^1 (their iter t-1), so it
        // is safe to re-stage into it below.
        __syncthreads();

        if (t + 1 < ITERS)
            issue_tile(t + 1, cur ^ 1);               // overlaps the compute below

        const float* bf = (const float*)&stage[cur][0];           // stride-3 dwords:
        const int*   ti = (const int*)&stage[cur][BARY_BYTES];    // conflict-free (gcd(3,64)=1)
        const int tri = ti[tid];
        float w0 = bf[tid * 3 + 0];
        float w1 = bf[tid * 3 + 1];
        float w2 = bf[tid * 3 + 2];

        const bool  bg  = tri < 0;
        const float vis = bg ? 0.0f : 1.0f;
        // Branchless background: zero weights => zero result; clamped gather @0 is harmless.
        w0 *= vis; w1 *= vis; w2 *= vis;
        const int idx = bg ? 0 : tri;

        // Gather 3x16 f32 vertex attrs: 12x global_load_b128 from the L2-resident
        // table (7.7 MB << 192 MB L2).
        const float4* __restrict__ v =
            reinterpret_cast<const float4*>(attributes + (size_t)idx * 48);
        float4 r[4];
#pragma unroll
        for (int q = 0; q < 4; ++q) {
            const float4 a = v[q], b = v[q + 4], c = v[q + 8];
            r[q].x = fmaf(w2, c.x, fmaf(w1, b.x, w0 * a.x));
            r[q].y = fmaf(w2, c.y, fmaf(w1, b.y, w0 * a.y));
            r[q].z = fmaf(w2, c.z, fmaf(w1, b.z, w0 * a.z));
            r[q].w = fmaf(w2, c.w, fmaf(w1, b.w, w0 * a.w));
        }

        // [B, 17, H, W] output: per-channel b32 stores are coalesced across lanes.
        // Non-temporal: output is write-once; keep L2 for the attribute table.
        const int p    = t * span + pbase + tid;
        const int bimg = p >> 18;                     // p / (H*W)
        const int pix  = p & (HWPIX - 1);
        float* ob = out + (size_t)bimg * (DCH + 1) * HWPIX + pix;

#pragma unroll
        for (int q = 0; q < 4; ++q) {
            __builtin_nontemporal_store(r[q].x, ob + (size_t)(4 * q + 0) * HWPIX);
            __builtin_nontemporal_store(r[q].y, ob + (size_t)(4 * q + 1) * HWPIX);
            __builtin_nontemporal_store(r[q].z, ob + (size_t)(4 * q + 2) * HWPIX);
            __builtin_nontemporal_store(r[q].w, ob + (size_t)(4 * q + 3) * HWPIX);
        }
        __builtin_nontemporal_store(vis, ob + (size_t)DCH * HWPIX);
    }
}

extern "C" void kernel_launch(void* const* d_in, const int* in_sizes, int n_in,
                              void* d_out, int out_size, void* d_ws, size_t ws_size,
                              hipStream_t stream) {
    const float* attributes = (const float*)d_in[0];   // [4, 10000, 3, 16] f32
    const int*   tribuf     = (const int*)  d_in[1];   // [4, 512, 512]     i32
    const float* barybuf    = (const float*)d_in[2];   // [4, 512, 512, 3]  f32
    float*       out        = (float*)d_out;           // [4, 17, 512, 512] f32

    raster_interp_kernel<<<BLOCKS, TPB, 0, stream>>>(attributes, tribuf, barybuf, out);
}